// GAN4LS_65549790871889
// MI455X (gfx1250) — compile-verified
//
#include <hip/hip_runtime.h>
#include <hip/hip_bf16.h>

// ---------------- problem constants (from reference) ----------------
#define NNODES 50000
#define NEDGES 640000
#define ETOT   (NEDGES + NNODES)   // edges + self loops = 690000
#define FIN    768
#define FHID   128
#define NHEAD  4
#define CHAN   32
#define NEG_SLOPE 0.2f
#define MTILES (NNODES / 16)       // 3125
#define MPAIRS ((MTILES + 1) / 2)  // 1563
#define ROWS_PER_BLK ((NNODES + 255) / 256)  // 196

typedef __attribute__((ext_vector_type(2))) float v2f;
typedef __attribute__((ext_vector_type(8))) float v8f;

// ---------------- helpers ----------------
__device__ __forceinline__ void atomicMaxF(float* addr, float value) {
    // order-preserving float max via int bit tricks (init must be -inf)
    if (value >= 0.0f)
        atomicMax((int*)addr, __float_as_int(value));
    else
        atomicMin((unsigned int*)addr, __float_as_uint(value));
}

// ---------------- fp32 WMMA GEMM: XL = X@Wl, XR = X@Wr ----------------
// One wave per PAIR of 16-row tiles of X. The 32 B-fragment loads per K-step
// (shared weights) now feed 32 WMMAs (2 M-tiles x 8 N-tiles x {Wl,Wr}),
// halving the VMEM:WMMA instruction ratio vs one tile per wave. x is streamed
// from HBM exactly once; W stays cache-resident. A/B fragment layout per
// CDNA5 ISA 7.12.2 (16x4 f32). Wave-uniform exits/guards keep EXEC all-1s.
__global__ __launch_bounds__(256) void k_gemm_xlxr(
    const float* __restrict__ X, int K,
    const float* __restrict__ Wl, const float* __restrict__ Wr,
    float* __restrict__ XL, float* __restrict__ XR)
{
    const int pair = (int)((blockIdx.x * blockDim.x + threadIdx.x) >> 5);
    if (pair >= MPAIRS) return;                 // whole-wave exit
    const int lane = threadIdx.x & 31;
    const int half = lane >> 4;                 // 0: K=0..1, 1: K=2..3
    const int l16  = lane & 15;
    const int mt0  = pair * 2;
    const int mt1  = (mt0 + 1 < MTILES) ? (mt0 + 1) : (MTILES - 1); // clamp tail
    const int m0   = mt0 * 16;
    const int m1   = mt1 * 16;

    v8f accL0[8], accR0[8], accL1[8], accR1[8];
    const v8f vz = {};
#pragma unroll
    for (int i = 0; i < 8; ++i) {
        accL0[i] = vz; accR0[i] = vz; accL1[i] = vz; accR1[i] = vz;
    }

    const float* xrow0 = X + (size_t)(m0 + l16) * K;
    const float* xrow1 = X + (size_t)(m1 + l16) * K;
    for (int k0 = 0; k0 < K; k0 += 4) {
        const int kk = k0 + 2 * half;
        v2f a0, a1;
        a0.x = xrow0[kk];
        a0.y = xrow0[kk + 1];
        a1.x = xrow1[kk];
        a1.y = xrow1[kk + 1];
#pragma unroll
        for (int nt = 0; nt < 8; ++nt) {
            const int col = nt * 16 + l16;
            v2f bl, br;
            bl.x = Wl[(size_t)kk * FHID + col];
            bl.y = Wl[(size_t)(kk + 1) * FHID + col];
            br.x = Wr[(size_t)kk * FHID + col];
            br.y = Wr[(size_t)(kk + 1) * FHID + col];
            accL0[nt] = __builtin_amdgcn_wmma_f32_16x16x4_f32(
                false, a0, false, bl, (short)0, accL0[nt], false, false);
            accL1[nt] = __builtin_amdgcn_wmma_f32_16x16x4_f32(
                false, a1, false, bl, (short)0, accL1[nt], false, false);
            accR0[nt] = __builtin_amdgcn_wmma_f32_16x16x4_f32(
                false, a0, false, br, (short)0, accR0[nt], false, false);
            accR1[nt] = __builtin_amdgcn_wmma_f32_16x16x4_f32(
                false, a1, false, br, (short)0, accR1[nt], false, false);
        }
    }
    // D layout: VGPR r -> rows m+r (lanes 0-15) and m+r+8 (lanes 16-31)
#pragma unroll
    for (int nt = 0; nt < 8; ++nt) {
#pragma unroll
        for (int r = 0; r < 8; ++r) {
            const int row = m0 + r + 8 * half;
            const int col = nt * 16 + l16;
            XL[(size_t)row * FHID + col] = accL0[nt][r];
            XR[(size_t)row * FHID + col] = accR0[nt][r];
        }
    }
    if (mt0 + 1 < MTILES) {                     // wave-uniform tail guard
#pragma unroll
        for (int nt = 0; nt < 8; ++nt) {
#pragma unroll
            for (int r = 0; r < 8; ++r) {
                const int row = m1 + r + 8 * half;
                const int col = nt * 16 + l16;
                XL[(size_t)row * FHID + col] = accL1[nt][r];
                XR[(size_t)row * FHID + col] = accR1[nt][r];
            }
        }
    }
}

// ---------------- per-layer scratch init (float4 on agg) ----------------
__global__ void k_init(float4* __restrict__ agg4, float* __restrict__ m,
                       float* __restrict__ s)
{
    const size_t idx = (size_t)blockIdx.x * blockDim.x + threadIdx.x;
    if (idx < (size_t)NNODES * (FHID / 4)) {
        agg4[idx] = make_float4(0.f, 0.f, 0.f, 0.f);
    }
    if (idx < (size_t)NNODES * NHEAD) {
        s[idx] = 0.0f;
        m[idx] = __int_as_float(0xFF800000u);   // -inf
    }
}

// ---------------- edge pass 1: e = att . leakyrelu(xl[src]+xr[dst]); segmax
// one wave per edge; one b128 per lane covers the whole 128-float row.
// lane l owns channels 4l..4l+3 -> head h = l>>3; per-head reduce = 3 shfl_xor
// within 8-lane groups.
__global__ __launch_bounds__(256) void k_edge_max(
    const float* __restrict__ XL, const float* __restrict__ XR,
    const float* __restrict__ att, const int* __restrict__ ei,
    float* __restrict__ ebuf, float* __restrict__ m)
{
    const int e = (int)((blockIdx.x * blockDim.x + threadIdx.x) >> 5);
    if (e >= ETOT) return;
    const int lane = threadIdx.x & 31;
    int src, dst;
    if (e < NEDGES) { src = ei[e]; dst = ei[NEDGES + e]; }
    else            { src = e - NEDGES; dst = src; }       // self loop
    const float4 vl = ((const float4*)(XL + (size_t)src * FHID))[lane];
    const float4 vr = ((const float4*)(XR + (size_t)dst * FHID))[lane];
    const float4 va = ((const float4*)att)[lane];

    float t, v;
    t = vl.x + vr.x; t = (t > 0.f) ? t : NEG_SLOPE * t; v  = t * va.x;
    t = vl.y + vr.y; t = (t > 0.f) ? t : NEG_SLOPE * t; v += t * va.y;
    t = vl.z + vr.z; t = (t > 0.f) ? t : NEG_SLOPE * t; v += t * va.z;
    t = vl.w + vr.w; t = (t > 0.f) ? t : NEG_SLOPE * t; v += t * va.w;
    v += __shfl_xor(v, 4, 32);
    v += __shfl_xor(v, 2, 32);
    v += __shfl_xor(v, 1, 32);
    if ((lane & 7) == 0) {
        const int h = lane >> 3;
        ebuf[(size_t)e * NHEAD + h] = v;
        atomicMaxF(&m[dst * NHEAD + h], v);
    }
}

// ---------------- edge pass 2: a = exp(e - m[dst]); s += a; agg += a*xl[src]
__global__ __launch_bounds__(256) void k_edge_acc(
    const float* __restrict__ XL, const int* __restrict__ ei,
    const float* __restrict__ ebuf, const float* __restrict__ m,
    float* __restrict__ s, float* __restrict__ agg)
{
    const int e = (int)((blockIdx.x * blockDim.x + threadIdx.x) >> 5);
    if (e >= ETOT) return;
    const int lane = threadIdx.x & 31;
    int src, dst;
    if (e < NEDGES) { src = ei[e]; dst = ei[NEDGES + e]; }
    else            { src = e - NEDGES; dst = src; }
    const int h = lane >> 3;   // 8-lane groups share a head (broadcast loads)
    const float a = __expf(ebuf[(size_t)e * NHEAD + h] - m[dst * NHEAD + h]);
    if ((lane & 7) == 0) atomicAdd(&s[dst * NHEAD + h], a);
    const float4 vl = ((const float4*)(XL + (size_t)src * FHID))[lane];
    float* pa = agg + (size_t)dst * FHID + lane * 4;
    atomicAdd(pa + 0, a * vl.x);
    atomicAdd(pa + 1, a * vl.y);
    atomicAdd(pa + 2, a * vl.z);
    atomicAdd(pa + 3, a * vl.w);
}

// ---------------- node finalize: out = agg/(s+eps) + bias, optional relu ----
__global__ void k_finalize(const float4* __restrict__ agg4,
                           const float* __restrict__ s,
                           const float4* __restrict__ bias4,
                           float4* __restrict__ out4, int relu)
{
    const size_t idx = (size_t)blockIdx.x * blockDim.x + threadIdx.x;
    if (idx >= (size_t)NNODES * (FHID / 4)) return;
    const int node = (int)(idx >> 5);
    const int q    = (int)(idx & 31);       // float4 slot within row
    const int h    = q >> 3;
    const float inv = 1.0f / (s[node * NHEAD + h] + 1e-16f);
    float4 v = agg4[idx];
    const float4 bb = bias4[q];
    v.x = v.x * inv + bb.x;
    v.y = v.y * inv + bb.y;
    v.z = v.z * inv + bb.z;
    v.w = v.w * inv + bb.w;
    if (relu) {
        v.x = fmaxf(v.x, 0.f); v.y = fmaxf(v.y, 0.f);
        v.z = fmaxf(v.z, 0.f); v.w = fmaxf(v.w, 0.f);
    }
    out4[idx] = v;
}

// ---------------- mean stage 1: coalesced per-block column partials --------
__global__ __launch_bounds__(256) void k_colpart(const float* __restrict__ hbuf,
                                                 float* __restrict__ partial)
{
    const int b   = blockIdx.x;            // 0..255
    const int t   = threadIdx.x;
    const int col = t & 127;
    const int sub = t >> 7;                // 0 or 1
    const int r0 = b * ROWS_PER_BLK;
    int r1 = r0 + ROWS_PER_BLK;
    if (r1 > NNODES) r1 = NNODES;
    float sum = 0.0f;
    for (int r = r0 + sub; r < r1; r += 2)
        sum += hbuf[(size_t)r * FHID + col];
    __shared__ float red[256];
    red[t] = sum;
    __syncthreads();
    if (t < 128) partial[b * FHID + col] = red[t] + red[t + 128];
}

// ---------------- mean stage 2 + classifier: out = mean(h) @ clfW + clfb ----
__global__ void k_clf(const float* __restrict__ partial,
                      const float* __restrict__ W,
                      const float* __restrict__ b, float* __restrict__ out)
{
    const int t = threadIdx.x;             // 128 threads
    float sum = 0.0f;
    for (int blk = 0; blk < 256; ++blk)
        sum += partial[blk * FHID + t];
    const float gmean = sum / (float)NNODES;
    __shared__ float p0[128], p1[128];
    p0[t] = gmean * W[t * 2 + 0];
    p1[t] = gmean * W[t * 2 + 1];
    __syncthreads();
    for (int o = 64; o > 0; o >>= 1) {
        if (t < o) { p0[t] += p0[t + o]; p1[t] += p1[t + o]; }
        __syncthreads();
    }
    if (t == 0) { out[0] = p0[0] + b[0]; out[1] = p1[0] + b[1]; }
}

// ---------------- host side ----------------
extern "C" void kernel_launch(void* const* d_in, const int* in_sizes, int n_in,
                              void* d_out, int out_size, void* d_ws, size_t ws_size,
                              hipStream_t stream)
{
    const float* x   = (const float*)d_in[0];
    const int*   ei  = (const int*)d_in[1];
    const float *Wl[4], *Wr[4], *att[4], *bias[4];
    for (int i = 0; i < 4; ++i) {
        Wl[i]   = (const float*)d_in[2 + 4 * i];
        Wr[i]   = (const float*)d_in[3 + 4 * i];
        att[i]  = (const float*)d_in[4 + 4 * i];
        bias[i] = (const float*)d_in[5 + 4 * i];
    }
    const float* clfW = (const float*)d_in[18];
    const float* clfb = (const float*)d_in[19];
    float* out = (float*)d_out;

    // workspace carve-up (floats)
    float* p    = (float*)d_ws;
    float* XL   = p; p += (size_t)NNODES * FHID;   // 6.4M
    float* XR   = p; p += (size_t)NNODES * FHID;
    float* AGG  = p; p += (size_t)NNODES * FHID;
    float* HB   = p; p += (size_t)NNODES * FHID;
    float* M    = p; p += (size_t)NNODES * NHEAD;
    float* S    = p; p += (size_t)NNODES * NHEAD;
    float* EB   = p; p += (size_t)ETOT * NHEAD;
    float* PART = p; p += 256 * FHID;

    const dim3 blk(256);
    const dim3 gemmGrid((MPAIRS + 7) / 8);                  // 8 waves / block
    const dim3 nodeGrid4((NNODES * (FHID / 4) + 255) / 256);// float4 granularity
    const dim3 edgeGrid((ETOT + 7) / 8);                    // 8 edges / block

    for (int layer = 0; layer < 4; ++layer) {
        const float* in = (layer == 0) ? x : HB;
        const int    K  = (layer == 0) ? FIN : FHID;
        hipLaunchKernelGGL(k_gemm_xlxr, gemmGrid, blk, 0, stream,
                           in, K, Wl[layer], Wr[layer], XL, XR);
        hipLaunchKernelGGL(k_init, nodeGrid4, blk, 0, stream,
                           (float4*)AGG, M, S);
        hipLaunchKernelGGL(k_edge_max, edgeGrid, blk, 0, stream,
                           XL, XR, att[layer], ei, EB, M);
        hipLaunchKernelGGL(k_edge_acc, edgeGrid, blk, 0, stream,
                           XL, ei, EB, M, S, AGG);
        hipLaunchKernelGGL(k_finalize, nodeGrid4, blk, 0, stream,
                           (const float4*)AGG, S, (const float4*)bias[layer],
                           (float4*)HB, (layer < 3) ? 1 : 0);
    }
    hipLaunchKernelGGL(k_colpart, dim3(256), blk, 0, stream, HB, PART);
    hipLaunchKernelGGL(k_clf, dim3(1), dim3(128), 0, stream,
                       PART, clfW, clfb, out);
}